// Seq2Seq_65489661329789
// MI455X (gfx1250) — compile-verified
//
#include <hip/hip_runtime.h>
#include <hip/hip_bf16.h>

// ---------------------------------------------------------------------------
// Seq2Seq forward on MI455X (gfx1250, wave32).
// All GEMMs are X[M,K] @ W[N,K]^T -> v_wmma_f32_16x16x4_f32 with fp32 acc.
// Attention collapses analytically (softmax over size-1 axis == 1 -> ctx=enc0).
// Encoder (256 steps) / decoder (32 steps) are persistent kernels with a
// monotonic atomic grid barrier across 12 workgroups (96 waves).
// Data movement: waves within a WG share the A block (streams x from HBM
// once, ~268MB ~= 12us @ 23.3TB/s); logits output (270MB) stored NT to keep
// Wlog/sent resident in the 192MB L2.
// ---------------------------------------------------------------------------

typedef float v2f __attribute__((ext_vector_type(2)));
typedef float v8f __attribute__((ext_vector_type(8)));

#define Hh   256
#define Tt   256
#define Bb   64
#define Ll   33
#define Vv   32000
#define INF_ 4096
#define G3   768            // 3*H
#define BOSI 1
#define NWG  12             // persistent workgroups (96 waves)

// ---------------- wave-level 32x32 GEMM block: C += A @ W^T ----------------
template <bool NTA>
__device__ __forceinline__ void wave_gemm32(const float* __restrict__ A, int lda, int m0,
                                            const float* __restrict__ W, int ldw, int n0,
                                            int K, int lane, v8f acc[2][2]) {
  const int l15  = lane & 15;
  const int koff = (lane >> 4) << 1;          // 0 or 2
  const float* a0 = A + (size_t)(m0 + l15)      * lda + koff;
  const float* a1 = A + (size_t)(m0 + 16 + l15) * lda + koff;
  const float* w0 = W + (size_t)(n0 + l15)      * ldw + koff;
  const float* w1 = W + (size_t)(n0 + 16 + l15) * ldw + koff;
#pragma unroll 4
  for (int k = 0; k < K; k += 4) {
    v2f av0, av1;
    if (NTA) {
      av0 = __builtin_nontemporal_load((const v2f*)(a0 + k));
      av1 = __builtin_nontemporal_load((const v2f*)(a1 + k));
    } else {
      av0 = *(const v2f*)(a0 + k);
      av1 = *(const v2f*)(a1 + k);
    }
    v2f wv0 = *(const v2f*)(w0 + k);
    v2f wv1 = *(const v2f*)(w1 + k);
    acc[0][0] = __builtin_amdgcn_wmma_f32_16x16x4_f32(false, av0, false, wv0, (short)0, acc[0][0], false, false);
    acc[0][1] = __builtin_amdgcn_wmma_f32_16x16x4_f32(false, av0, false, wv1, (short)0, acc[0][1], false, false);
    acc[1][0] = __builtin_amdgcn_wmma_f32_16x16x4_f32(false, av1, false, wv0, (short)0, acc[1][0], false, false);
    acc[1][1] = __builtin_amdgcn_wmma_f32_16x16x4_f32(false, av1, false, wv1, (short)0, acc[1][1], false, false);
  }
}

// C/D layout: VGPR v -> row = v + 8*(lane>>4), col = lane&15.
__device__ __forceinline__ void wave_store32(float* __restrict__ C, int ldc, int m0, int n0,
                                             const float* __restrict__ bias, int lane,
                                             const v8f acc[2][2]) {
  const int l15 = lane & 15, half = lane >> 4;
#pragma unroll
  for (int ni = 0; ni < 2; ++ni) {
    const int col = n0 + ni * 16 + l15;
    const float bv = bias ? bias[col] : 0.f;
#pragma unroll
    for (int mi = 0; mi < 2; ++mi) {
#pragma unroll
      for (int v = 0; v < 8; ++v) {
        const int row = m0 + mi * 16 + half * 8 + v;
        C[(size_t)row * ldc + col] = acc[mi][ni][v] + bv;
      }
    }
  }
}

// Transposed NT store for logits: GEMM row r = l*B + b -> out[b][l][col].
__device__ __forceinline__ void wave_store_logits(float* __restrict__ out, int m0, int n0,
                                                  const float* __restrict__ bias, int lane,
                                                  const v8f acc[2][2]) {
  const int l15 = lane & 15, half = lane >> 4;
#pragma unroll
  for (int ni = 0; ni < 2; ++ni) {
    const int col = n0 + ni * 16 + l15;
    const float bv = bias[col];
#pragma unroll
    for (int mi = 0; mi < 2; ++mi) {
#pragma unroll
      for (int v = 0; v < 8; ++v) {
        const int row = m0 + mi * 16 + half * 8 + v;
        const int l = row >> 6, b = row & 63;        // row = l*64 + b
        __builtin_nontemporal_store(acc[mi][ni][v] + bv,
                                    &out[((size_t)b * Ll + l) * Vv + col]);
      }
    }
  }
}

// ---------------- generic GEMM kernel (one 32x32 block per wave) -----------
// Job mapping: bn = job % nbn, bm = job / nbn  ->  the 8 waves of a WG share
// one A block (A streamed from HBM once; W blocks served from L2).
template <int MODE, bool NTA>
__global__ void __launch_bounds__(256) gemm_xwT_kernel(const float* __restrict__ A, int lda,
                                                       const float* __restrict__ W, int ldw,
                                                       const float* __restrict__ bias,
                                                       float* __restrict__ C, int ldc,
                                                       int M, int N, int K) {
  const int lane = threadIdx.x & 31;
  const int wid  = (blockIdx.x << 3) | (threadIdx.x >> 5);
  const int nw   = gridDim.x << 3;
  const int nbn  = N >> 5;
  const int njobs = (M >> 5) * nbn;
  for (int job = wid; job < njobs; job += nw) {
    const int bn = job % nbn, bm = job / nbn;
    v8f acc[2][2] = {};
    wave_gemm32<NTA>(A, lda, bm << 5, W, ldw, bn << 5, K, lane, acc);
    if (MODE == 0) wave_store32(C, ldc, bm << 5, bn << 5, bias, lane, acc);
    else           wave_store_logits(C, bm << 5, bn << 5, bias, lane, acc);
  }
}

// ---------------- monotonic atomic grid barrier ----------------------------
__device__ __forceinline__ void gbar(unsigned* ctr, unsigned target) {
  __syncthreads();
  __threadfence();
  if (threadIdx.x == 0) {
    __atomic_fetch_add(ctr, 1u, __ATOMIC_SEQ_CST);
    while (__atomic_load_n(ctr, __ATOMIC_SEQ_CST) < target)
      __builtin_amdgcn_s_sleep(2);
  }
  __threadfence();
  __syncthreads();
}

__device__ __forceinline__ float sigf(float x) { return 1.f / (1.f + expf(-x)); }

// ---------------- persistent encoder: 256 GRU steps ------------------------
__global__ void __launch_bounds__(256) encoder_kernel(
    const float* __restrict__ xe, const float* __restrict__ Wi1, const float* __restrict__ bi1,
    const float* __restrict__ Wh1, const float* __restrict__ bh1,
    float* __restrict__ giws, float* __restrict__ ghws,
    float* __restrict__ hA, float* __restrict__ hB,
    float* __restrict__ enc0, float* __restrict__ hlast, unsigned* __restrict__ ctr) {
  const int lane = threadIdx.x & 31;
  const int wid  = (blockIdx.x << 3) | (threadIdx.x >> 5);   // 0..95
  unsigned bc = 0;
  for (int t = 0; t < Tt; ++t) {
    const float* h  = (t & 1) ? hB : hA;
    float*       hn = (t & 1) ? hA : hB;
    if (wid < 48) {
      const int bm = wid & 1, bn = wid >> 1;                 // 2 x 24 blocks
      v8f acc[2][2] = {};
      wave_gemm32<false>(h, Hh, bm << 5, Wh1, Hh, bn << 5, Hh, lane, acc);
      wave_store32(ghws, G3, bm << 5, bn << 5, bh1, lane, acc);
    } else {
      const int j = wid - 48, bm = j & 1, bn = j >> 1;
      v8f acc[2][2] = {};
      wave_gemm32<false>(xe + (size_t)t * Hh, Tt * Hh, bm << 5, Wi1, Hh, bn << 5, Hh, lane, acc);
      wave_store32(giws, G3, bm << 5, bn << 5, bi1, lane, acc);
    }
    gbar(ctr, ++bc * NWG);
    if (wid < Bb) {
      const int b = wid;
#pragma unroll
      for (int j = 0; j < 8; ++j) {
        const int c = lane + (j << 5);
        const float r  = sigf(giws[b * G3 + c]            + ghws[b * G3 + c]);
        const float z  = sigf(giws[b * G3 + Hh + c]       + ghws[b * G3 + Hh + c]);
        const float nc = tanhf(giws[b * G3 + 2 * Hh + c] + r * ghws[b * G3 + 2 * Hh + c]);
        const float h2 = (1.f - z) * nc + z * h[b * Hh + c];
        hn[b * Hh + c] = h2;
        if (t == 0)      enc0[b * Hh + c]  = h2;   // outs[0]
        if (t == Tt - 1) hlast[b * Hh + c] = h2;
      }
    }
    gbar(ctr, ++bc * NWG);
  }
}

// ---------------- persistent decoder: 32 steps -----------------------------
__global__ void __launch_bounds__(256) decoder_kernel(
    const float* __restrict__ emb, const float* __restrict__ Wi2, const float* __restrict__ bi2,
    const float* __restrict__ Wh2, const float* __restrict__ bh2,
    const float* __restrict__ enc0, const float* __restrict__ hlast,
    float* __restrict__ sent, float* __restrict__ gictx, float* __restrict__ g1ws,
    float* __restrict__ gh2ws, float* __restrict__ diws, unsigned* __restrict__ ctr) {
  const int lane = threadIdx.x & 31;
  const int wid  = (blockIdx.x << 3) | (threadIdx.x >> 5);
  const int tid  = blockIdx.x * 256 + threadIdx.x;
  unsigned bc = 0;
  // P0: gictx = enc0 @ Wi2[:,H:]^T + bi2 (ctx == enc0 since singleton softmax == 1)
  if (wid < 48) {
    const int bm = wid & 1, bn = wid >> 1;
    v8f acc[2][2] = {};
    wave_gemm32<false>(enc0, Hh, bm << 5, Wi2 + Hh, 2 * Hh, bn << 5, Hh, lane, acc);
    wave_store32(gictx, G3, bm << 5, bn << 5, bi2, lane, acc);
  }
  for (int e = tid; e < Bb * Hh; e += NWG * 256) {           // di0 = sent[0] = emb[BOS]
    const float v = emb[BOSI * Hh + (e & (Hh - 1))];
    diws[e] = v;
    sent[e] = v;
  }
  gbar(ctr, ++bc * NWG);

  for (int s = 0; s < Ll - 1; ++s) {
    const float* hp = (s == 0) ? hlast : (sent + (size_t)s * Bb * Hh);
    if (wid < 48) {
      const int bm = wid & 1, bn = wid >> 1;
      v8f acc[2][2] = {};
      wave_gemm32<false>(diws, Hh, bm << 5, Wi2, 2 * Hh, bn << 5, Hh, lane, acc);
      wave_store32(g1ws, G3, bm << 5, bn << 5, nullptr, lane, acc);
    } else {
      const int j = wid - 48, bm = j & 1, bn = j >> 1;
      v8f acc[2][2] = {};
      wave_gemm32<false>(hp, Hh, bm << 5, Wh2, Hh, bn << 5, Hh, lane, acc);
      wave_store32(gh2ws, G3, bm << 5, bn << 5, bh2, lane, acc);
    }
    gbar(ctr, ++bc * NWG);
    if (wid < Bb) {
      const int b = wid;
      float bv = -3.402823466e38f; int bidx = 0;
#pragma unroll
      for (int j = 0; j < 8; ++j) {
        const int c = lane + (j << 5);
        const float gr = g1ws[b * G3 + c]            + gictx[b * G3 + c];
        const float gz = g1ws[b * G3 + Hh + c]       + gictx[b * G3 + Hh + c];
        const float gn = g1ws[b * G3 + 2 * Hh + c]   + gictx[b * G3 + 2 * Hh + c];
        const float r  = sigf(gr + gh2ws[b * G3 + c]);
        const float z  = sigf(gz + gh2ws[b * G3 + Hh + c]);
        const float nc = tanhf(gn + r * gh2ws[b * G3 + 2 * Hh + c]);
        const float h2 = (1.f - z) * nc + z * hp[b * Hh + c];
        sent[(size_t)(s + 1) * Bb * Hh + b * Hh + c] = h2;
        if (h2 > bv) { bv = h2; bidx = c; }                  // ascending c -> first max
      }
      for (int off = 16; off; off >>= 1) {                   // wave32 argmax reduce
        const float ov = __shfl_xor(bv, off, 32);
        const int   oi = __shfl_xor(bidx, off, 32);
        if (ov > bv || (ov == bv && oi < bidx)) { bv = ov; bidx = oi; }
      }
#pragma unroll
      for (int j = 0; j < 8; ++j) {
        const int c = lane + (j << 5);
        diws[b * Hh + c] = emb[(size_t)bidx * Hh + c];
      }
    }
    gbar(ctr, ++bc * NWG);
  }
}

// ---------------------------------------------------------------------------
extern "C" void kernel_launch(void* const* d_in, const int* in_sizes, int n_in,
                              void* d_out, int out_size, void* d_ws, size_t ws_size,
                              hipStream_t stream) {
  (void)in_sizes; (void)n_in; (void)out_size; (void)ws_size;
  const float* x    = (const float*)d_in[0];
  const float* W_in = (const float*)d_in[1];
  const float* b_in = (const float*)d_in[2];
  const float* Wi1  = (const float*)d_in[3];
  const float* Wh1  = (const float*)d_in[4];
  const float* bi1  = (const float*)d_in[5];
  const float* bh1  = (const float*)d_in[6];
  const float* Wi2  = (const float*)d_in[7];
  const float* Wh2  = (const float*)d_in[8];
  const float* bi2  = (const float*)d_in[9];
  const float* bh2  = (const float*)d_in[10];
  // d_in[11..16] = Wa/ba/Ua/bua/Va/bva: dead code (singleton softmax == 1)
  const float* emb  = (const float*)d_in[17];
  const float* Wlog = (const float*)d_in[18];
  const float* blog = (const float*)d_in[19];
  float* out = (float*)d_out;

  unsigned* ctr = (unsigned*)d_ws;                 // [0]=encoder barrier, [1]=decoder
  float* fws  = (float*)((char*)d_ws + 256);
  float* xe   = fws;                               // 16384 x 256
  float* ghws = xe + (size_t)Bb * Tt * Hh;         // 64 x 768
  float* giws = ghws + Bb * G3;                    // 64 x 768
  float* hA   = giws + Bb * G3;                    // 64 x 256
  float* hB   = hA + Bb * Hh;
  float* enc0 = hB + Bb * Hh;
  float* hl   = enc0 + Bb * Hh;
  float* sent = hl + Bb * Hh;                      // 33 x 64 x 256
  float* gictx = sent + (size_t)Ll * Bb * Hh;      // 64 x 768
  float* g1   = gictx + Bb * G3;
  float* gh2  = g1 + Bb * G3;
  float* di   = gh2 + Bb * G3;                     // 64 x 256

  hipMemsetAsync(d_ws, 0, 256, stream);                           // barrier counters
  hipMemsetAsync(hA, 0, (size_t)Bb * Hh * sizeof(float), stream); // h0 = 0

  // 1) xe = x @ W_inlin^T + b  (M=16384, N=256, K=4096); NT x loads,
  //    WG-shared A blocks -> x streamed from HBM once.
  gemm_xwT_kernel<0, true><<<512, 256, 0, stream>>>(x, INF_, W_in, INF_, b_in,
                                                    xe, Hh, Bb * Tt, Hh, INF_);
  // 2) encoder GRU (persistent, 12 WGs)
  encoder_kernel<<<NWG, 256, 0, stream>>>(xe, Wi1, bi1, Wh1, bh1,
                                          giws, ghws, hA, hB, enc0, hl, ctr);
  // 3) decoder (persistent, 12 WGs)
  decoder_kernel<<<NWG, 256, 0, stream>>>(emb, Wi2, bi2, Wh2, bh2, enc0, hl,
                                          sent, gictx, g1, gh2, di, ctr + 1);
  // 4) logits = sent @ Wlog^T + blog, NT-stored transposed to [B,L,V]
  gemm_xwT_kernel<1, false><<<8250, 256, 0, stream>>>(sent, Hh, Wlog, Hh, blog,
                                                      out, Vv, Ll * Bb, Vv, Hh);
}